// Renderer_53523882443250
// MI455X (gfx1250) — compile-verified
//
#include <hip/hip_runtime.h>
#include <hip/hip_bf16.h>
#include <math.h>

// Problem constants (match reference)
#define HH 512
#define WW 512
#define NS 5000      // strokes
#define SP 10        // bezier samples per stroke
#define KK 20        // K nearest
#define GC 51        // coarse grid per dim (int(512*0.1))
#define NQ (GC*GC)   // 2601 coarse queries
#define QT ((NQ+15)/16)  // 163 query tiles
#define ST ((NS+15)/16)  // 313 stroke tiles

typedef float v2f __attribute__((ext_vector_type(2)));
typedef float v8f __attribute__((ext_vector_type(8)));

// ---------------------------------------------------------------------------
// Kernel 1: per-stroke prep.
//   cp[i][j]  : bezier points (uses RAW location, like reference)
//   feat[i]   : (-2*lx_clip, -2*ly_clip, lx_clip^2+ly_clip^2, 0)  [WMMA B operand]
//   colw[i]   : (r,g,b clipped to [0,1], width)
// ---------------------------------------------------------------------------
__global__ __launch_bounds__(256)
void prep_kernel(const float* __restrict__ cs, const float* __restrict__ ce,
                 const float* __restrict__ cc, const float* __restrict__ lo,
                 const float* __restrict__ co, const float* __restrict__ wi,
                 float* __restrict__ cp, float4* __restrict__ feat,
                 float4* __restrict__ colw) {
  int i = blockIdx.x * blockDim.x + threadIdx.x;
  if (i >= NS) return;
  float lx = lo[2*i], ly = lo[2*i+1];
  float sx = cs[2*i] + lx, sy = cs[2*i+1] + ly;
  float ex = ce[2*i] + lx, ey = ce[2*i+1] + ly;
  float cx = cc[2*i] + lx, cy = cc[2*i+1] + ly;
#pragma unroll
  for (int j = 0; j < SP; ++j) {
    float t    = (float)j * (1.0f/9.0f);
    float omt  = 1.0f - t;
    float omt2 = omt*omt, t2 = t*t;
    cp[i*2*SP + 2*j + 0] = cx + omt2*(sx - cx) + t2*(ex - cx);
    cp[i*2*SP + 2*j + 1] = cy + omt2*(sy - cy) + t2*(ey - cy);
  }
  float lcx = fminf(fmaxf(lx, 0.0f), (float)WW);
  float lcy = fminf(fmaxf(ly, 0.0f), (float)HH);
  feat[i] = make_float4(-2.0f*lcx, -2.0f*lcy, lcx*lcx + lcy*lcy, 0.0f);
  colw[i] = make_float4(fminf(fmaxf(co[3*i+0], 0.0f), 1.0f),
                        fminf(fmaxf(co[3*i+1], 0.0f), 1.0f),
                        fminf(fmaxf(co[3*i+2], 0.0f), 1.0f),
                        wi[i]);
}

// ---------------------------------------------------------------------------
// Kernel 2: fused WMMA distance matrix + top-20 selection.
// One 128-thread workgroup per 16-query tile. Each of the 4 waves computes one
// 16x16 d2 tile per iteration with V_WMMA_F32_16X16X4_F32:
//   d2 = qx*(-2lx) + qy*(-2ly) + (qx^2+qy^2)*1 + 1*(lx^2+ly^2)
// A (16x4 f32, 2 VGPRs): lanes 0-15 = {K0=qx, K1=qy}, lanes 16-31 = {K2=|q|^2, K3=1}
// B (4x16 f32, 2 VGPRs): lanes 0-15 = {K0=-2lx, K1=-2ly}, lanes 16-31 = {K2=1, K3=|l|^2}
// C/D (16x16 f32, 8 VGPRs): reg r, lane L -> M = r + 8*(L>=16), N = L&15
// EXEC stays full across the WMMA (operand build is branchless selects).
// Selection scans strokes in ascending id order with strict-< insertion, which
// matches jax.lax.top_k's "lower index first on ties".
// ---------------------------------------------------------------------------
__global__ __launch_bounds__(128)
void knn_kernel(const float4* __restrict__ feat, int* __restrict__ knn) {
  __shared__ float d2buf[4][16][16];
  const int tid   = threadIdx.x;
  const int lane  = tid & 31;
  const int wv    = tid >> 5;
  const int half  = lane >> 4;      // 0: K0/K1 half, 1: K2/K3 half
  const int mrow  = lane & 15;
  const int qbase = blockIdx.x * 16;

  // A operand (loop invariant). Query r = i*51+j -> x = t51[i], y = t51[j].
  int qi = qbase + mrow; qi = qi < NQ ? qi : NQ - 1;
  float qx = (float)(qi / GC) * (512.0f/50.0f);
  float qy = (float)(qi % GC) * (512.0f/50.0f);
  v2f a;
  a.x = half ? (qx*qx + qy*qy) : qx;
  a.y = half ? 1.0f            : qy;

  // per-selector-lane sorted top-20 (ascending d2, ties keep lower stroke id)
  float bv[KK]; int bi[KK];
#pragma unroll
  for (int k = 0; k < KK; ++k) { bv[k] = 3.4e38f; bi[k] = 0; }

  for (int base = 0; base < ST; base += 4) {
    int tile = base + wv; tile = tile < ST ? tile : ST - 1;
    int n = tile * 16 + mrow; n = n < NS ? n : NS - 1;
    float4 f = feat[n];
    v2f b;
    b.x = half ? 1.0f : f.x;
    b.y = half ? f.z  : f.y;
    v8f c = {0.f,0.f,0.f,0.f,0.f,0.f,0.f,0.f};
    c = __builtin_amdgcn_wmma_f32_16x16x4_f32(
        /*neg_a=*/false, a, /*neg_b=*/false, b,
        /*c_mod=*/(short)0, c, /*reuse_a=*/false, /*reuse_b=*/false);
#pragma unroll
    for (int r = 0; r < 8; ++r)
      d2buf[wv][r + half*8][mrow] = c[r];
    __syncthreads();

    if (tid < 16) {   // 16 selector lanes, one per query row of the tile
      for (int w = 0; w < 4; ++w) {
        int t2 = base + w;
        if (t2 >= ST) break;
#pragma unroll
        for (int nn = 0; nn < 16; ++nn) {
          int sid = t2 * 16 + nn;
          if (sid < NS) {
            float v = d2buf[w][tid][nn];
            if (v < bv[KK-1]) {          // strict <: ties reject newer (higher id)
              bv[KK-1] = v; bi[KK-1] = sid;
#pragma unroll
              for (int p = KK-1; p > 0; --p) {
                if (bv[p] < bv[p-1]) {
                  float tv = bv[p]; bv[p] = bv[p-1]; bv[p-1] = tv;
                  int   ti = bi[p]; bi[p] = bi[p-1]; bi[p-1] = ti;
                }
              }
            }
          }
        }
      }
    }
    __syncthreads();
  }

  if (tid < 16) {
    int q = qbase + tid;
    if (q < NQ) {
#pragma unroll
      for (int k = 0; k < KK; ++k) knn[q*KK + k] = bi[k];
    }
  }
}

// ---------------------------------------------------------------------------
// Kernel 3: per-pixel render. Reproduces the reference's transpose quirks:
//   B_a / colors / widths from coarse cell (m(u), m(v))
//   B_ba               from coarse cell (m(v), m(u))
//   nearest = B_ba * (1 + t)   (reference bug, preserved)
// Stroke tables (~660 KB) are L2-resident (192 MB L2), gathers are cheap.
// ---------------------------------------------------------------------------
__global__ __launch_bounds__(256)
void render_kernel(const float* __restrict__ cp, const float4* __restrict__ colw,
                   const int* __restrict__ knn, float* __restrict__ out) {
  int pid = blockIdx.x * 256 + threadIdx.x;   // exactly 512*512 threads
  int u = pid >> 9;        // row (h)
  int v = pid & 511;       // col (w)
  float px = (float)v * (512.0f/511.0f);
  float py = (float)u * (512.0f/511.0f);
  int mu = (u * GC) >> 9;  // u*51/512
  int mv = (v * GC) >> 9;
  const int* ka = knn + (mu*GC + mv) * KK;   // B_a / color / width side
  const int* kb = knn + (mv*GC + mu) * KK;   // B_ba side (transposed quirk)
  const float2* cp2 = (const float2*)cp;

  float xk[KK]; int ia[KK];
  float D = 3.4e38f;
  float xmax = -3.4e38f;
  for (int k = 0; k < KK; ++k) {
    int iA = ka[k]; int iB = kb[k];
    ia[k] = iA;
    const float2* pa = cp2 + iA * SP;
    const float2* pb = cp2 + iB * SP;
    if (k + 1 < KK) __builtin_prefetch(cp2 + kb[k+1] * SP, 0, 0);
    float dk = 3.4e38f;
    float2 bp = pb[0];
#pragma unroll
    for (int s = 0; s < SP-1; ++s) {
      float2 bn = pb[s+1];
      float2 as = pa[s];
      float bax = bn.x - bp.x, bay = bn.y - bp.y;
      float pbx = px - as.x,  pby = py - as.y;
      float tt = (bax*pbx + bay*pby) / (bax*bax + bay*bay + 1e-10f);
      tt = fminf(fmaxf(tt, 0.0f), 1.0f);
      float nx = bax + tt*bax, ny = bay + tt*bay;   // reference bug preserved
      float dx = px - nx, dy = py - ny;
      dk = fminf(dk, dx*dx + dy*dy);
      bp = bn;
    }
    D = fminf(D, dk);
    float x = 100000.0f / (1e-8f + dk);
    xk[k] = x;
    xmax = fmaxf(xmax, x);
  }
  // softmax over K + weighted color/width + sigmoid blend
  float esum = 0.0f, cr = 0.0f, cg = 0.0f, cb = 0.0f, bw = 0.0f;
  for (int k = 0; k < KK; ++k) {
    float e = expf(xk[k] - xmax);
    float4 cwv = colw[ia[k]];
    esum += e;
    cr += cwv.x * e; cg += cwv.y * e; cb += cwv.z * e; bw += cwv.w * e;
  }
  float inv = 1.0f / esum;
  cr *= inv; cg *= inv; cb *= inv; bw *= inv;
  float bm = 1.0f / (1.0f + expf(-(bw - D)));
  float bg = (1.0f - bm) * 0.5f;
  int o = pid * 3;
  out[o+0] = cr * bm + bg;
  out[o+1] = cg * bm + bg;
  out[o+2] = cb * bm + bg;
}

// ---------------------------------------------------------------------------
// Workspace layout (floats from d_ws base; all offsets 16B aligned):
//   [0      .. 100000) cp    : 5000*10*2 bezier points
//   [100000 .. 120000) feat  : 5000 float4 WMMA-B features
//   [120000 .. 140000) colw  : 5000 float4 (rgb + width)
//   [140000 .. +52020) knn   : 2601*20 int indices
// Total ~768 KB. Everything is fully rewritten each call (deterministic).
// ---------------------------------------------------------------------------
extern "C" void kernel_launch(void* const* d_in, const int* in_sizes, int n_in,
                              void* d_out, int out_size, void* d_ws, size_t ws_size,
                              hipStream_t stream) {
  (void)in_sizes; (void)n_in; (void)out_size; (void)ws_size;
  const float* cs = (const float*)d_in[0];  // curve_s (5000,2)
  const float* ce = (const float*)d_in[1];  // curve_e (5000,2)
  const float* cc = (const float*)d_in[2];  // curve_c (5000,2)
  const float* lo = (const float*)d_in[3];  // location (5000,2)
  const float* co = (const float*)d_in[4];  // color (5000,3)
  const float* wi = (const float*)d_in[5];  // width (5000,1)

  float*  ws   = (float*)d_ws;
  float*  cp   = ws;
  float4* feat = (float4*)(ws + 100000);
  float4* colw = (float4*)(ws + 120000);
  int*    knn  = (int*)(ws + 140000);
  float*  out  = (float*)d_out;

  hipLaunchKernelGGL(prep_kernel, dim3((NS + 255)/256), dim3(256), 0, stream,
                     cs, ce, cc, lo, co, wi, cp, feat, colw);
  hipLaunchKernelGGL(knn_kernel, dim3(QT), dim3(128), 0, stream, feat, knn);
  hipLaunchKernelGGL(render_kernel, dim3((HH*WW)/256), dim3(256), 0, stream,
                     cp, colw, knn, out);
}